// GeoFlowNet_70025146794439
// MI455X (gfx1250) — compile-verified
//
#include <hip/hip_runtime.h>
#include <hip/hip_bf16.h>

// ============================================================================
// GeoFlowNet forward on gfx1250 (MI455X). fp32 end-to-end, fp32 WMMA
// (v_wmma_f32_16x16x4_f32) for both the pointwise-MLP GEMMs and the
// Gaussian-interpolation (normalized-kernel) N x M x C contractions.
// Network is latency-bound (≈5 GFLOP total, ~10 MB working set => L2-resident),
// so the design goals are: branchless WMMA inner loops, LDS-resident FPS with
// shuffle-based argmax, and prefetch-overlapped LDS staging in gauss.
// ============================================================================

typedef __attribute__((ext_vector_type(2))) float v2f;
typedef __attribute__((ext_vector_type(8))) float v8f;

#define WMMA_F32_4(a, b, c) \
  __builtin_amdgcn_wmma_f32_16x16x4_f32(false, (a), false, (b), (short)0, (c), false, false)

// ---------------------------------------------------------------------------
// ChannelConv GEMM: Y[N, Cout] = X[N, Cin] @ W[Cin, Cout] + bias (+ReLU).
// One wave computes a full 16 x (NT*16) row-block: A fragment loaded once per
// K-step, reused across NT WMMAs. Main K loop is guard-free; the ragged tail
// (Cin % 4 != 0) uses clamped unconditional loads + selects (no exec churn).
// GN=true adds a column clamp/store-guard for Cout not a multiple of 16.
// Fragment layouts per CDNA5 ISA:
//   A (16x4):  lane m = l&15, K = 2*(l>>4) + v
//   B (4x16):  lane n = l&15, K = 2*(l>>4) + v
//   C/D:       row M = r + 8*(l>>4), col N = l&15
// ---------------------------------------------------------------------------
template <int NT, bool GN>
__global__ __launch_bounds__(32) void cc_gemm_t(
    const float* __restrict__ X, int ldx,
    const float* __restrict__ W, const float* __restrict__ bias,
    float* __restrict__ Y, int ldy,
    int Cin, int Cout, int relu)
{
  const int lane = threadIdx.x;
  const int h = lane >> 4;
  const int t = lane & 15;
  const int m0 = blockIdx.x << 4;
  const float* xrow = X + (size_t)(m0 + t) * ldx;

  v8f acc[NT];
#pragma unroll
  for (int i = 0; i < NT; ++i) acc[i] = (v8f){};

  const int kmain = Cin & ~3;
  for (int k0 = 0; k0 < kmain; k0 += 4) {
    const int ka = k0 + 2 * h;
    v2f a;
    a.x = xrow[ka];
    a.y = xrow[ka + 1];
    const float* w0 = W + (size_t)ka * Cout;
#pragma unroll
    for (int nt = 0; nt < NT; ++nt) {
      int n = nt * 16 + t;
      if (GN) n = (n < Cout) ? n : (Cout - 1);   // clamped; junk cols dropped at store
      v2f b;
      b.x = w0[n];
      b.y = w0[Cout + n];
      acc[nt] = WMMA_F32_4(a, b, acc[nt]);
    }
  }
  if (kmain < Cin) {                              // ragged K tail (Cin % 4 != 0)
    const int ka  = kmain + 2 * h;
    const int k0c = (ka     < Cin) ? ka     : (Cin - 1);
    const int k1c = (ka + 1 < Cin) ? ka + 1 : (Cin - 1);
    const float ax = xrow[k0c];                   // unconditional clamped loads
    const float ay = xrow[k1c];
    v2f a;
    a.x = (ka     < Cin) ? ax : 0.0f;
    a.y = (ka + 1 < Cin) ? ay : 0.0f;
    const float* w0c = W + (size_t)k0c * Cout;
    const float* w1c = W + (size_t)k1c * Cout;
#pragma unroll
    for (int nt = 0; nt < NT; ++nt) {
      int n = nt * 16 + t;
      if (GN) n = (n < Cout) ? n : (Cout - 1);
      const float bx = w0c[n];
      const float by = w1c[n];
      v2f b;
      b.x = (ka     < Cin) ? bx : 0.0f;
      b.y = (ka + 1 < Cin) ? by : 0.0f;
      acc[nt] = WMMA_F32_4(a, b, acc[nt]);
    }
  }

#pragma unroll
  for (int nt = 0; nt < NT; ++nt) {
    const int n = nt * 16 + t;
    if (GN && n >= Cout) continue;                // post-WMMA divergence is fine
    const float bv = bias[n];
#pragma unroll
    for (int r = 0; r < 8; ++r) {
      const int row = m0 + r + 8 * h;
      float v = acc[nt][r] + bv;
      if (relu) v = fmaxf(v, 0.0f);
      Y[(size_t)row * ldy + n] = v;
    }
  }
}

// ---------------------------------------------------------------------------
// BatchNorm training statistics: mean/var per channel over all N rows.
// One block per channel. meanvar = [mean[0..C), var[0..C)].
// ---------------------------------------------------------------------------
__global__ __launch_bounds__(256) void bn_stats(
    const float* __restrict__ Y, int ldy, int N, int Cout,
    float* __restrict__ meanvar)
{
  const int c = blockIdx.x;
  __shared__ float ss[256];
  __shared__ float sq[256];
  float s = 0.0f, q = 0.0f;
  for (int i = threadIdx.x; i < N; i += 256) {
    const float v = Y[(size_t)i * ldy + c];
    s += v;
    q += v * v;
  }
  ss[threadIdx.x] = s;
  sq[threadIdx.x] = q;
  __syncthreads();
  for (int w = 128; w > 0; w >>= 1) {
    if ((int)threadIdx.x < w) {
      ss[threadIdx.x] += ss[threadIdx.x + w];
      sq[threadIdx.x] += sq[threadIdx.x + w];
    }
    __syncthreads();
  }
  if (threadIdx.x == 0) {
    const float m = ss[0] / (float)N;
    meanvar[c]        = m;
    meanvar[Cout + c] = sq[0] / (float)N - m * m;   // biased var (ddof=0)
  }
}

__global__ void bn_apply_relu(
    float* __restrict__ Y, int ldy, int N, int Cout,
    const float* __restrict__ gamma, const float* __restrict__ beta,
    const float* __restrict__ meanvar)
{
  const int idx = blockIdx.x * 256 + threadIdx.x;
  if (idx >= N * Cout) return;
  const int r = idx / Cout;
  const int c = idx - r * Cout;
  float y = Y[(size_t)r * ldy + c];
  y = (y - meanvar[c]) * rsqrtf(meanvar[Cout + c] + 1e-5f) * gamma[c] + beta[c];
  Y[(size_t)r * ldy + c] = fmaxf(y, 0.0f);
}

// ---------------------------------------------------------------------------
// Furthest point sampling — the serial critical path of the network
// (2048+512+128 dependent argmax rounds per cloud). Everything (coords +
// running min-distances) lives in dynamic LDS (4*N floats <= 64 KB of the
// 320 KB WGP pool). Argmax: wave32 shuffle butterfly + 32-entry second stage
// => 3 barriers per round instead of log2(1024)+1. Tie-break = smallest index
// (matches jnp.argmax). Fully deterministic.
// ---------------------------------------------------------------------------
#define FPS_T 1024
__global__ __launch_bounds__(FPS_T) void fps_kernel(
    const float* __restrict__ pts, float* __restrict__ out, int N, int npoint)
{
  extern __shared__ float smem[];
  float* sd = smem;                 // [N]
  float* sx = sd + N;               // [N]
  float* sy = sx + N;               // [N]
  float* sz = sy + N;               // [N]
  float* rv = sz + N;               // [32]
  int*   ri = (int*)(rv + 32);      // [32]

  const int t    = threadIdx.x;
  const int lane = t & 31;
  const int wid  = t >> 5;

  for (int j = t; j < N; j += FPS_T) {
    sx[j] = pts[3 * j];
    sy[j] = pts[3 * j + 1];
    sz[j] = pts[3 * j + 2];
    sd[j] = 3.4e38f;
  }
  if (t == 0) { out[0] = pts[0]; out[1] = pts[1]; out[2] = pts[2]; }
  __syncthreads();

  int cur = 0;
  for (int i = 1; i < npoint; ++i) {
    const float lx = sx[cur], ly = sy[cur], lz = sz[cur];
    float bestv = -1.0f;
    int   besti = 0x7fffffff;
    for (int j = t; j < N; j += FPS_T) {
      const float dx = sx[j] - lx;
      const float dy = sy[j] - ly;
      const float dz = sz[j] - lz;
      const float dm = fminf(sd[j], dx * dx + dy * dy + dz * dz);
      sd[j] = dm;
      if (dm > bestv || (dm == bestv && j < besti)) { bestv = dm; besti = j; }
    }
    // wave32 butterfly argmax (max value, min index on ties)
#pragma unroll
    for (int off = 16; off > 0; off >>= 1) {
      const float ov = __shfl_xor(bestv, off, 32);
      const int   oi = __shfl_xor(besti, off, 32);
      if (ov > bestv || (ov == bestv && oi < besti)) { bestv = ov; besti = oi; }
    }
    if (lane == 0) { rv[wid] = bestv; ri[wid] = besti; }
    __syncthreads();
    if (wid == 0) {                         // reduce the 32 wave winners
      float v  = rv[lane];
      int   ii = ri[lane];
#pragma unroll
      for (int off = 16; off > 0; off >>= 1) {
        const float ov = __shfl_xor(v, off, 32);
        const int   oi = __shfl_xor(ii, off, 32);
        if (ov > v || (ov == v && oi < ii)) { v = ov; ii = oi; }
      }
      if (lane == 0) ri[0] = ii;
    }
    __syncthreads();
    cur = ri[0];
    if (t == 0) { out[3 * i] = sx[cur]; out[3 * i + 1] = sy[cur]; out[3 * i + 2] = sz[cur]; }
    __syncthreads();                        // protect ri[0] before next round
  }
}

// ---------------------------------------------------------------------------
// Gaussian-kernel interpolation:
//   O[n,:] = sum_m w(n,m) * F[m,:] / (sum_m w(n,m) + 1e-8),
//   w(n,m) = exp(-|q_n - s_m|^2 * inv2s2)
// 4 waves/block, each wave owns 16 query rows. Support coords+features staged
// through LDS in 64-point chunks; next chunk prefetched (global_prefetch_b8)
// while the current one is consumed. Weights are computed in-register as the
// WMMA A fragment (2 exps/lane/K-step); features come from LDS as B fragments.
// Row sums accumulate alongside; 1/(sum+1e-8) fused via wave32 shuffles.
// Requires Nq % 64 == 0, Ms % 64 == 0, C = CT*16 (true for this network).
// ---------------------------------------------------------------------------
#define GCH 64
template <int CT>
__global__ __launch_bounds__(128) void gauss_k(
    const float* __restrict__ Q, const float* __restrict__ S,
    const float* __restrict__ F, int ldf,
    float* __restrict__ O, int ldo, int Ms, float inv2s2)
{
  constexpr int C = CT * 16;
  __shared__ float sS[GCH * 3];
  __shared__ float sF[GCH * C];

  const int tid  = threadIdx.x;
  const int wave = tid >> 5;
  const int lane = tid & 31;
  const int h = lane >> 4;
  const int t = lane & 15;
  const int q0 = blockIdx.x * 64 + wave * 16;
  const int qr = q0 + t;
  const float qx = Q[3 * qr], qy = Q[3 * qr + 1], qz = Q[3 * qr + 2];

  v8f acc[CT];
#pragma unroll
  for (int i = 0; i < CT; ++i) acc[i] = (v8f){};
  float wsum = 0.0f;

  for (int s0 = 0; s0 < Ms; s0 += GCH) {
    __syncthreads();
    for (int i = tid; i < GCH * 3; i += 128) sS[i] = S[s0 * 3 + i];
    for (int i = tid; i < GCH * C; i += 128) {
      const int kk = i / C;
      const int c  = i - kk * C;
      sF[i] = F[(size_t)(s0 + kk) * ldf + c];
    }
    // speculative prefetch of the next chunk (lowers to global_prefetch_b8)
    const int snext = s0 + GCH;
    if (snext < Ms) {
      __builtin_prefetch((const char*)(F + (size_t)snext * ldf) + tid * 128, 0, 1);
      if (tid < 2) __builtin_prefetch((const char*)(S + snext * 3) + tid * 128, 0, 1);
    }
    __syncthreads();

#pragma unroll
    for (int ks = 0; ks < GCH; ks += 4) {
      const int k0 = ks + 2 * h;
      float dx = sS[3 * k0]     - qx;
      float dy = sS[3 * k0 + 1] - qy;
      float dz = sS[3 * k0 + 2] - qz;
      const float d20 = dx * dx + dy * dy + dz * dz;
      dx = sS[3 * k0 + 3] - qx;
      dy = sS[3 * k0 + 4] - qy;
      dz = sS[3 * k0 + 5] - qz;
      const float d21 = dx * dx + dy * dy + dz * dz;
      v2f a;
      a.x = __expf(-d20 * inv2s2);
      a.y = __expf(-d21 * inv2s2);
      wsum += a.x + a.y;
#pragma unroll
      for (int ct = 0; ct < CT; ++ct) {
        v2f b;
        b.x = sF[k0 * C + ct * 16 + t];
        b.y = sF[(k0 + 1) * C + ct * 16 + t];
        acc[ct] = WMMA_F32_4(a, b, acc[ct]);
      }
    }
  }

  wsum += __shfl_xor(wsum, 16, 32);            // lanes l and l+16 share row l&15
  const float inv = 1.0f / (wsum + 1e-8f);

#pragma unroll
  for (int ct = 0; ct < CT; ++ct) {
#pragma unroll
    for (int r = 0; r < 8; ++r) {
      const int row = r + 8 * h;               // 0..15
      const float invr = __shfl(inv, row, 32); // lane 'row' holds its 1/sum
      O[(size_t)(q0 + row) * ldo + ct * 16 + t] = acc[ct][r] * invr;
    }
  }
}

// ---------------------------------------------------------------------------
// Strided block copy (materializes concatenations as column slices).
// ---------------------------------------------------------------------------
__global__ void copy_block(const float* __restrict__ src, int lds_,
                           float* __restrict__ dst, int ldd, int total, int C)
{
  const int i = blockIdx.x * 256 + threadIdx.x;
  if (i >= total) return;
  const int r = i / C;
  const int c = i - r * C;
  dst[(size_t)r * ldd + c] = src[(size_t)r * lds_ + c];
}

// ===========================================================================
// Host orchestration
// ===========================================================================
namespace {

struct CCp {
  const float *w, *b, *g, *be;
  int cin, cout, bn;
};

// _SPECS order
static const int kCin[26]  = {1, 16, 16, 16, 19, 32, 32, 32, 35, 64, 64, 128, 64,
                              67, 64, 128, 192, 96, 96, 128, 64, 64, 80, 32, 32, 32};
static const int kCout[26] = {16, 16, 16, 16, 32, 32, 32, 32, 64, 64, 64, 64, 64,
                              64, 64, 64, 96, 96, 96, 64, 64, 64, 32, 32, 32, 3};
static const int kBn[26]   = {0, 0, 1, 0, 0, 1, 0, 1, 0, 1, 0, 0, 1,
                              0, 1, 0, 0, 1, 1, 0, 1, 1, 0, 1, 1, 0};

static void run_cc(const CCp& p, const float* X, int ldx, float* Y, int N,
                   float* meanvar, hipStream_t st)
{
  const int relu = p.bn ? 0 : 1;
  dim3 g(N / 16), b(32);
  switch (p.cout) {
    case 16:
      cc_gemm_t<1, false><<<g, b, 0, st>>>(X, ldx, p.w, p.b, Y, p.cout, p.cin, p.cout, relu);
      break;
    case 32:
      cc_gemm_t<2, false><<<g, b, 0, st>>>(X, ldx, p.w, p.b, Y, p.cout, p.cin, p.cout, relu);
      break;
    case 64:
      cc_gemm_t<4, false><<<g, b, 0, st>>>(X, ldx, p.w, p.b, Y, p.cout, p.cin, p.cout, relu);
      break;
    case 96:
      cc_gemm_t<6, false><<<g, b, 0, st>>>(X, ldx, p.w, p.b, Y, p.cout, p.cin, p.cout, relu);
      break;
    default:
      cc_gemm_t<1, true><<<g, b, 0, st>>>(X, ldx, p.w, p.b, Y, p.cout, p.cin, p.cout, relu);
      break;
  }
  if (p.bn) {
    bn_stats<<<dim3(p.cout), dim3(256), 0, st>>>(Y, p.cout, N, p.cout, meanvar);
    const int tot = N * p.cout;
    bn_apply_relu<<<dim3((tot + 255) / 256), dim3(256), 0, st>>>(
        Y, p.cout, N, p.cout, p.g, p.be, meanvar);
  }
}

static void run_gauss(const float* Q, const float* S, const float* F, int ldf,
                      float* O, int ldo, int Nq, int Ms, int C, float inv2s2,
                      hipStream_t st)
{
  dim3 g(Nq / 64), b(128);
  switch (C) {
    case 16: gauss_k<1><<<g, b, 0, st>>>(Q, S, F, ldf, O, ldo, Ms, inv2s2); break;
    case 32: gauss_k<2><<<g, b, 0, st>>>(Q, S, F, ldf, O, ldo, Ms, inv2s2); break;
    case 64: gauss_k<4><<<g, b, 0, st>>>(Q, S, F, ldf, O, ldo, Ms, inv2s2); break;
    case 96: gauss_k<6><<<g, b, 0, st>>>(Q, S, F, ldf, O, ldo, Ms, inv2s2); break;
    default: break;
  }
}

static void run_fps(const float* pts, float* out, int N, int npoint, hipStream_t st)
{
  const size_t shmem = (size_t)(4 * N) * sizeof(float) + 64 * sizeof(float);
  fps_kernel<<<dim3(1), dim3(FPS_T), shmem, st>>>(pts, out, N, npoint);
}

static void run_copy(const float* s, int lds_, float* d, int ldd, int N, int C,
                     hipStream_t st)
{
  const int tot = N * C;
  copy_block<<<dim3((tot + 255) / 256), dim3(256), 0, st>>>(s, lds_, d, ldd, tot, C);
}

}  // namespace

extern "C" void kernel_launch(void* const* d_in, const int* in_sizes, int n_in,
                              void* d_out, int out_size, void* d_ws, size_t ws_size,
                              hipStream_t stream)
{
  (void)in_sizes; (void)n_in; (void)out_size; (void)ws_size;

  // --- inputs (setup_inputs() dict order) ---
  const float* pc1 = (const float*)d_in[0];   // [4096,3]
  const float* pc2 = (const float*)d_in[1];   // [4096,3]
  const float* fe1 = (const float*)d_in[2];   // [4096,1]
  const float* fe2 = (const float*)d_in[3];   // [4096,1]

  // params: for each spec (insertion order): w, b, [gamma, beta]
  CCp cc[26];
  int pi = 4;
  for (int s = 0; s < 26; ++s) {
    cc[s].cin = kCin[s]; cc[s].cout = kCout[s]; cc[s].bn = kBn[s];
    cc[s].w = (const float*)d_in[pi++];
    cc[s].b = (const float*)d_in[pi++];
    if (cc[s].bn) {
      cc[s].g  = (const float*)d_in[pi++];
      cc[s].be = (const float*)d_in[pi++];
    } else {
      cc[s].g = nullptr; cc[s].be = nullptr;
    }
  }

  // --- workspace bump allocator (≈ 11 MB total) ---
  char* wp = (char*)d_ws;
  auto alloc = [&](size_t nfl) -> float* {
    float* p = (float*)wp;
    wp += ((nfl * sizeof(float) + 255) / 256) * 256;
    return p;
  };

  float* A16   = alloc(4096 * 16);
  float* B16   = alloc(4096 * 16);
  float* G16   = alloc(4096 * 16);
  float* L0F1  = alloc(4096 * 16);
  float* L0F2  = alloc(4096 * 16);
  float* L1P1  = alloc(2048 * 3);
  float* L1P2  = alloc(2048 * 3);
  float* CAT19 = alloc(2048 * 19);
  float* A32   = alloc(2048 * 32);
  float* B32   = alloc(2048 * 32);
  float* G32   = alloc(2048 * 32);
  float* L1F1  = alloc(2048 * 32);
  float* L1F2  = alloc(2048 * 32);
  float* L2P1  = alloc(512 * 3);
  float* L2P2  = alloc(512 * 3);
  float* CAT35 = alloc(512 * 35);
  float* A64   = alloc(512 * 64);
  float* G64   = alloc(512 * 64);
  float* L2F1_ = alloc(512 * 64);
  float* L2F2_ = alloc(512 * 64);
  float* L2F1  = alloc(512 * 64);
  float* L2F2  = alloc(512 * 64);
  float* CAT128= alloc(512 * 128);
  float* L3P1  = alloc(128 * 3);
  float* L3P2  = alloc(128 * 3);
  float* CAT67 = alloc(128 * 67);
  float* A3    = alloc(128 * 64);
  float* L3F1_ = alloc(128 * 64);
  float* L3F2_ = alloc(128 * 64);
  float* L3CAT = alloc(128 * 128);
  float* L3O   = alloc(128 * 64);
  float* DCAT  = alloc(512 * 192);
  float* A96   = alloc(512 * 96);
  float* B96   = alloc(512 * 96);
  float* L2FF  = alloc(512 * 96);
  float* L1CAT = alloc(2048 * 128);
  float* A64b  = alloc(2048 * 64);
  float* B64b  = alloc(2048 * 64);
  float* L1FF  = alloc(2048 * 64);
  float* L0CAT = alloc(4096 * 80);
  float* A32b  = alloc(4096 * 32);
  float* B32b  = alloc(4096 * 32);
  float* L0FF  = alloc(4096 * 32);
  float* MV    = alloc(2 * 96);

  // sigmas: S0=0.25, S1=1.0, S2=2.0, S3=4.0 -> 1/(2*sigma^2)
  const float IS0 = 8.0f, IS1 = 0.5f, IS2 = 0.125f, IS3 = 0.03125f;
  hipStream_t st = stream;

  // ---------------- level 0 encoders ----------------
  for (int cl = 0; cl < 2; ++cl) {
    const float* pc = cl ? pc2 : pc1;
    const float* fe = cl ? fe2 : fe1;
    float* l0f = cl ? L0F2 : L0F1;
    run_cc(cc[0], fe, 1, A16, 4096, MV, st);
    run_cc(cc[1], A16, 16, B16, 4096, MV, st);
    run_gauss(pc, pc, B16, 16, G16, 16, 4096, 4096, 16, IS0, st);
    run_cc(cc[2], G16, 16, A16, 4096, MV, st);
    run_cc(cc[3], A16, 16, l0f, 4096, MV, st);
  }

  // ---------------- level 1 ----------------
  run_fps(pc1, L1P1, 4096, 2048, st);
  run_fps(pc2, L1P2, 4096, 2048, st);
  for (int cl = 0; cl < 2; ++cl) {
    const float* pc = cl ? pc2 : pc1;
    const float* l0f = cl ? L0F2 : L0F1;
    float* l1p = cl ? L1P2 : L1P1;
    float* l1f = cl ? L1F2 : L1F1;
    run_gauss(l1p, pc, l0f, 16, CAT19, 19, 2048, 4096, 16, IS1, st);
    run_copy(l1p, 3, CAT19 + 16, 19, 2048, 3, st);
    run_cc(cc[4], CAT19, 19, A32, 2048, MV, st);
    run_cc(cc[5], A32, 32, B32, 2048, MV, st);
    run_gauss(l1p, l1p, B32, 32, G32, 32, 2048, 2048, 32, IS1, st);
    run_cc(cc[6], G32, 32, A32, 2048, MV, st);
    run_cc(cc[7], A32, 32, l1f, 2048, MV, st);
  }

  // ---------------- level 2 (cross-cloud) ----------------
  run_fps(L1P1, L2P1, 2048, 512, st);
  run_fps(L1P2, L2P2, 2048, 512, st);
  for (int cl = 0; cl < 2; ++cl) {
    const float* l1p = cl ? L1P2 : L1P1;
    const float* l1f = cl ? L1F2 : L1F1;
    float* l2p = cl ? L2P2 : L2P1;
    float* l2f_ = cl ? L2F2_ : L2F1_;
    run_gauss(l2p, l1p, l1f, 32, CAT35, 35, 512, 2048, 32, IS2, st);
    run_copy(l2p, 3, CAT35 + 32, 35, 512, 3, st);
    run_cc(cc[8], CAT35, 35, A64, 512, MV, st);
    run_cc(cc[9], A64, 64, l2f_, 512, MV, st);
  }
  run_gauss(L2P1, L2P2, L2F2_, 64, G64, 64, 512, 512, 64, IS2, st);
  run_cc(cc[10], G64, 64, A64, 512, MV, st);
  run_copy(A64, 64, CAT128, 128, 512, 64, st);
  run_copy(L2F1_, 64, CAT128 + 64, 128, 512, 64, st);
  run_cc(cc[11], CAT128, 128, L2F1, 512, MV, st);
  run_cc(cc[12], L2F2_, 64, L2F2, 512, MV, st);

  // ---------------- level 3 (cross-cloud) ----------------
  run_fps(L2P1, L3P1, 512, 128, st);
  run_fps(L2P2, L3P2, 512, 128, st);
  for (int cl = 0; cl < 2; ++cl) {
    const float* l2p = cl ? L2P2 : L2P1;
    const float* l2f = cl ? L2F2 : L2F1;
    float* l3p = cl ? L3P2 : L3P1;
    float* l3f_ = cl ? L3F2_ : L3F1_;
    run_gauss(l3p, l2p, l2f, 64, CAT67, 67, 128, 512, 64, IS3, st);
    run_copy(l3p, 3, CAT67 + 64, 67, 128, 3, st);
    run_cc(cc[13], CAT67, 67, A3, 128, MV, st);
    run_cc(cc[14], A3, 64, l3f_, 128, MV, st);
  }
  run_gauss(L3P1, L3P2, L3F2_, 64, L3CAT, 128, 128, 128, 64, IS3, st);
  run_copy(L3F1_, 64, L3CAT + 64, 128, 128, 64, st);
  run_cc(cc[15], L3CAT, 128, L3O, 128, MV, st);  // shared module -> compute once

  // ---------------- decoder 3 -> 2 -> 1 -> 0 ----------------
  run_gauss(L2P1, L3P1, L3O, 64, DCAT, 192, 512, 128, 64, IS2, st);
  run_copy(DCAT, 192, DCAT + 64, 192, 512, 64, st);      // identical second branch
  run_copy(L2F1, 64, DCAT + 128, 192, 512, 64, st);
  run_cc(cc[16], DCAT, 192, A96, 512, MV, st);
  run_cc(cc[17], A96, 96, B96, 512, MV, st);
  run_cc(cc[18], B96, 96, L2FF, 512, MV, st);

  run_gauss(L1P1, L2P1, L2FF, 96, L1CAT, 128, 2048, 512, 96, IS1, st);
  run_copy(L1F1, 32, L1CAT + 96, 128, 2048, 32, st);
  run_cc(cc[19], L1CAT, 128, A64b, 2048, MV, st);
  run_cc(cc[20], A64b, 64, B64b, 2048, MV, st);
  run_cc(cc[21], B64b, 64, L1FF, 2048, MV, st);

  run_gauss(pc1, L1P1, L1FF, 64, L0CAT, 80, 4096, 2048, 64, IS0, st);
  run_copy(L0F1, 16, L0CAT + 64, 80, 4096, 16, st);
  run_cc(cc[22], L0CAT, 80, A32b, 4096, MV, st);
  run_cc(cc[23], A32b, 32, B32b, 4096, MV, st);
  run_cc(cc[24], B32b, 32, L0FF, 4096, MV, st);

  // refine: plain linear (no BN, no ReLU) -> flow [4096,3]
  cc_gemm_t<1, true><<<dim3(4096 / 16), dim3(32), 0, st>>>(
      L0FF, 32, cc[25].w, cc[25].b, (float*)d_out, 3, 32, 3, 0);
}